// AdvancedAutoInformerModel_13391708029372
// MI455X (gfx1250) — compile-verified
//
#include <hip/hip_runtime.h>
#include <hip/hip_bf16.h>
#include <math.h>

// ---------------------------------------------------------------------------
// AdvancedAutoInformer, MI455X (gfx1250) version.
//
// Exploits block-local attention (BS=10) + single-token readout: only tokens
// 1990..1999 (x rows 1988..2001) ever influence the output. Whole model
// becomes, per batch: conv(10x16->10x256) -> +PE -> GEMM 16x256x256 (folded
// trend/seas mix) -> 2x [GEMM 16x768x256 (QKV) -> 8-head 10x10 attention ->
// GEMM 16x256x256 (out proj)] -> LN+FC on token 9.
// All GEMMs use V_WMMA_F32_16X16X4_F32 (exact fp32).
// ---------------------------------------------------------------------------

typedef __attribute__((ext_vector_type(2))) float v2f;
typedef __attribute__((ext_vector_type(8))) float v8f;

#define DHID 256
#define QKVW 768

// ---------------- prep kernels (workspace build, tiny) ----------------------

// M[d][e] = (d==e) + sum_k trend_w[k,d,e] + sum_k seas_w[k,d,e]
// c[e]    = sum_k trend_b[k,e] + sum_k seas_b[k,e]
__global__ void prep_mix_kernel(const float* __restrict__ tw,
                                const float* __restrict__ sw,
                                const float* __restrict__ tb,
                                const float* __restrict__ sb,
                                float* __restrict__ M, float* __restrict__ c) {
  int i = blockIdx.x * 256 + threadIdx.x;
  if (i < 256 * 256) {
    int d = i >> 8, e = i & 255;
    float v = (d == e) ? 1.0f : 0.0f;
    v += tw[i] + tw[65536 + i] + tw[131072 + i];
    v += sw[i] + sw[65536 + i] + sw[131072 + i];
    M[i] = v;
  }
  if (i < 256) {
    c[i] = tb[i] + tb[256 + i] + tb[512 + i] + sb[i] + sb[256 + i] + sb[512 + i];
  }
}

// qkvT[l][d][e] = wqkv[l][e][d]   (2 x 256 x 768)   B-layout for WMMA
// woT [l][d][e] = wo  [l][e][d]   (2 x 256 x 256)
__global__ void prep_wT_kernel(const float* __restrict__ wqkv,
                               const float* __restrict__ wo,
                               float* __restrict__ qkvT,
                               float* __restrict__ woT) {
  int i = blockIdx.x * 256 + threadIdx.x;
  if (i < 2 * QKVW * DHID) {
    int l = i / (QKVW * DHID);
    int r = i % (QKVW * DHID);
    int d = r / QKVW;
    int e = r % QKVW;
    qkvT[i] = wqkv[l * QKVW * DHID + e * DHID + d];
  }
  if (i < 2 * DHID * DHID) {
    int l = i >> 16;
    int r = i & 65535;
    int d = r >> 8;
    int e = r & 255;
    woT[i] = wo[l * 65536 + e * DHID + d];
  }
}

// ---------------- WMMA GEMM helper ------------------------------------------
// C(16 x N) = A(16 x K, LDS) * B(K x N, global) + bias(N), D in LDS.
// 8 waves split N into 16-wide tiles. Uses V_WMMA_F32_16X16X4_F32.
// A frag (ISA 7.12.2, 32-bit A 16x4): lane m=lane&15, khalf=lane>>4,
//   vgpr0/1 = A[m][k0+2*khalf +0/1].
// B frag (mirrored): lane n=lane&15, vgpr0/1 = B[k0+2*khalf +0/1][n].
// C frag: vgpr r = row r + 8*(lane>>4), col = lane&15.
__device__ __forceinline__ void wmma_gemm(const float* __restrict__ A, int lda,
                                          const float* __restrict__ B, int ldb,
                                          const float* __restrict__ bias,
                                          float* __restrict__ D, int ldd,
                                          int N, int K) {
  const int lane = threadIdx.x & 31;
  const int wave = threadIdx.x >> 5;
  const int lm = lane & 15;
  const int lh = lane >> 4;
  const int tilesPerWave = (N >> 4) >> 3;  // 8 waves
  for (int tt = 0; tt < tilesPerWave; ++tt) {
    const int nbase = ((wave * tilesPerWave) + tt) << 4;
    v8f acc = {0.f, 0.f, 0.f, 0.f, 0.f, 0.f, 0.f, 0.f};
    const float* __restrict__ bcol = B + nbase + lm;
    const float* __restrict__ arow = A + lm * lda + 2 * lh;
    for (int k0 = 0; k0 < K; k0 += 4) {
      v2f a = *(const v2f*)(arow + k0);
      const int ka = k0 + 2 * lh;
      v2f b;
      b[0] = bcol[(size_t)ka * ldb];
      b[1] = bcol[(size_t)(ka + 1) * ldb];
      acc = __builtin_amdgcn_wmma_f32_16x16x4_f32(
          false, a, false, b, (short)0, acc, false, false);
    }
    const float bv = bias ? bias[nbase + lm] : 0.0f;
#pragma unroll
    for (int r = 0; r < 8; ++r) {
      D[(8 * lh + r) * ldd + nbase + lm] = acc[r] + bv;
    }
  }
}

// ---------------- main kernel: one workgroup per batch ----------------------
__global__ __launch_bounds__(256) void autoinformer_tail_kernel(
    const float* __restrict__ x,        // (32,2000,16)
    const float* __restrict__ w1,       // (1,16,256)
    const float* __restrict__ b1,
    const float* __restrict__ w3,       // (3,16,256)
    const float* __restrict__ b3,
    const float* __restrict__ w5,       // (5,16,256)
    const float* __restrict__ b5,
    const float* __restrict__ M,        // (256,256) folded mix
    const float* __restrict__ cvec,     // (256)
    const float* __restrict__ qkvT,     // (2,256,768)
    const float* __restrict__ bqkv,     // (2,768)
    const float* __restrict__ woT,      // (2,256,256)
    const float* __restrict__ bo,       // (2,256)
    const float* __restrict__ ln_g,
    const float* __restrict__ ln_b,
    const float* __restrict__ fc_w,     // (256,4)
    const float* __restrict__ fc_b,     // (4)
    float* __restrict__ out) {          // (32,4)
  extern __shared__ float smem[];
  float* h0 = smem;                    // 16*256
  float* h1 = h0 + 16 * DHID;          // 16*256
  float* qkvb = h1 + 16 * DHID;        // 16*768
  float* xs = qkvb + 16 * QKVW;        // 14*16

  const int b = blockIdx.x;
  const int tid = threadIdx.x;
  const int lane = tid & 31;
  const int wave = tid >> 5;

  // ---- Phase 1: stage x rows 1988..2001 (rows >=2000 are conv zero-pad) ----
  if (tid < 14 * 16) {
    int row = tid >> 4, f = tid & 15;
    int s = 1988 + row;
    xs[tid] = (s < 2000) ? x[(size_t)b * 2000 * 16 + (size_t)s * 16 + f] : 0.0f;
  }
  __syncthreads();

  // ---- conv1+conv3+conv5 + positional encoding for tokens 1990..1999 ------
  {
    const int d = tid;  // 0..255 output channel
    float acc[10];
    const float bsum = b1[d] + b3[d] + b5[d];
#pragma unroll
    for (int t = 0; t < 10; ++t) acc[t] = bsum;
    // k=1: x row t+2
    for (int f = 0; f < 16; ++f) {
      float wv = w1[f * DHID + d];
#pragma unroll
      for (int t = 0; t < 10; ++t) acc[t] += xs[(t + 2) * 16 + f] * wv;
    }
    // k=3: x row t+1+j
    for (int j = 0; j < 3; ++j)
      for (int f = 0; f < 16; ++f) {
        float wv = w3[(j * 16 + f) * DHID + d];
#pragma unroll
        for (int t = 0; t < 10; ++t) acc[t] += xs[(t + 1 + j) * 16 + f] * wv;
      }
    // k=5: x row t+j
    for (int j = 0; j < 5; ++j)
      for (int f = 0; f < 16; ++f) {
        float wv = w5[(j * 16 + f) * DHID + d];
#pragma unroll
        for (int t = 0; t < 10; ++t) acc[t] += xs[(t + j) * 16 + f] * wv;
      }
    // positional encoding: pe[s, 2i]=sin(s*f_i), pe[s,2i+1]=cos(s*f_i)
    const float freq = expf((float)(d & ~1) * (-9.210340371976184f / 256.0f));
#pragma unroll
    for (int t = 0; t < 10; ++t) {
      float ang = (float)(1990 + t) * freq;
      acc[t] += (d & 1) ? cosf(ang) : sinf(ang);
      h0[t * DHID + d] = acc[t];
    }
#pragma unroll
    for (int t = 10; t < 16; ++t) h0[t * DHID + d] = 0.0f;
  }
  __syncthreads();

  // ---- folded decomposition mix: h1 = h0 @ M + c --------------------------
  wmma_gemm(h0, DHID, M, DHID, cvec, h1, DHID, DHID, DHID);
  __syncthreads();

  float* hin = h1;
  float* hout = h0;

  for (int l = 0; l < 2; ++l) {
    // ---- QKV projection: qkvb = hin @ WqkvT + bqkv ------------------------
    wmma_gemm(hin, DHID, qkvT + (size_t)l * DHID * QKVW, QKVW,
              bqkv + l * QKVW, qkvb, QKVW, QKVW, DHID);
    __syncthreads();

    // ---- block-local attention: one head per wave, head_dim = 32 = lanes --
    {
      const int hh = wave;  // 8 waves == 8 heads
      const int cb = hh * 32;
      for (int i = 0; i < 10; ++i) {
        float s = -1e30f;
        if (lane < 10) {
          float accq = 0.0f;
          const float* qi = qkvb + i * QKVW + cb;
          const float* kj = qkvb + lane * QKVW + 256 + cb;
#pragma unroll
          for (int dd = 0; dd < 32; ++dd) accq += qi[dd] * kj[dd];
          s = accq * 0.17677669529663687f;  // 1/sqrt(32)
        }
        float mx = s;
        for (int off = 16; off > 0; off >>= 1)
          mx = fmaxf(mx, __shfl_xor(mx, off, 32));
        float p = (lane < 10) ? expf(s - mx) : 0.0f;
        float sum = p;
        for (int off = 16; off > 0; off >>= 1) sum += __shfl_xor(sum, off, 32);
        float a = p / sum;  // attn prob in lanes 0..9
        float o = 0.0f;
#pragma unroll
        for (int j = 0; j < 10; ++j)
          o += __shfl(a, j, 32) * qkvb[j * QKVW + 512 + cb + lane];
        qkvb[i * QKVW + cb + lane] = o;  // overwrite dead q region
      }
    }
    __syncthreads();

    // ---- output projection: hout = o @ WoT + bo ---------------------------
    wmma_gemm(qkvb, QKVW, woT + (size_t)l * DHID * DHID, DHID,
              bo + l * DHID, hout, DHID, DHID, DHID);
    __syncthreads();

    float* tmp = hin;
    hin = hout;
    hout = tmp;
  }

  // ---- LayerNorm(token 9) + FC --------------------------------------------
  {
    float* red = qkvb;        // scratch
    float* lnv = qkvb + 512;  // 256 values
    const float v = hin[9 * DHID + tid];
    red[tid] = v;
    __syncthreads();
    for (int s = 128; s > 0; s >>= 1) {
      if (tid < s) red[tid] += red[tid + s];
      __syncthreads();
    }
    const float mu = red[0] * (1.0f / 256.0f);
    __syncthreads();
    const float dv = v - mu;
    red[tid] = dv * dv;
    __syncthreads();
    for (int s = 128; s > 0; s >>= 1) {
      if (tid < s) red[tid] += red[tid + s];
      __syncthreads();
    }
    const float var = red[0] * (1.0f / 256.0f);
    __syncthreads();
    const float ln = dv / sqrtf(var + 1e-5f) * ln_g[tid] + ln_b[tid];
    lnv[tid] = ln;
    __syncthreads();
    if (tid < 4) {
      float s = fc_b[tid];
      for (int t = 0; t < 256; ++t) s += lnv[t] * fc_w[t * 4 + tid];
      out[b * 4 + tid] = s;
    }
  }
}

// ---------------- host launch -----------------------------------------------
extern "C" void kernel_launch(void* const* d_in, const int* in_sizes, int n_in,
                              void* d_out, int out_size, void* d_ws,
                              size_t ws_size, hipStream_t stream) {
  (void)in_sizes; (void)n_in; (void)out_size; (void)ws_size;
  const float* x       = (const float*)d_in[0];
  const float* conv_w1 = (const float*)d_in[1];
  const float* conv_b1 = (const float*)d_in[2];
  const float* conv_w3 = (const float*)d_in[3];
  const float* conv_b3 = (const float*)d_in[4];
  const float* conv_w5 = (const float*)d_in[5];
  const float* conv_b5 = (const float*)d_in[6];
  const float* trend_w = (const float*)d_in[7];
  const float* trend_b = (const float*)d_in[8];
  const float* seas_w  = (const float*)d_in[9];
  const float* seas_b  = (const float*)d_in[10];
  const float* wqkv    = (const float*)d_in[11];
  const float* bqkv    = (const float*)d_in[12];
  const float* wo      = (const float*)d_in[13];
  const float* bo      = (const float*)d_in[14];
  const float* ln_g    = (const float*)d_in[15];
  const float* ln_b    = (const float*)d_in[16];
  const float* fc_w    = (const float*)d_in[17];
  const float* fc_b    = (const float*)d_in[18];

  float* ws   = (float*)d_ws;
  float* M    = ws;                    // 65536
  float* cvec = M + 65536;             // 256
  float* qkvT = cvec + 256;            // 2*256*768 = 393216
  float* woT  = qkvT + 393216;         // 2*256*256 = 131072

  prep_mix_kernel<<<256, 256, 0, stream>>>(trend_w, seas_w, trend_b, seas_b,
                                           M, cvec);
  prep_wT_kernel<<<1536, 256, 0, stream>>>(wqkv, wo, qkvT, woT);

  const size_t smemBytes =
      (size_t)(16 * DHID * 2 + 16 * QKVW + 14 * 16) * sizeof(float);
  autoinformer_tail_kernel<<<32, 256, smemBytes, stream>>>(
      x, conv_w1, conv_b1, conv_w3, conv_b3, conv_w5, conv_b5, M, cvec, qkvT,
      bqkv, woT, bo, ln_g, ln_b, fc_w, fc_b, (float*)d_out);
}